// Attention_48653389529249
// MI455X (gfx1250) — compile-verified
//
#include <hip/hip_runtime.h>
#include <hip/hip_bf16.h>
#include <math.h>

#define EMBED 256
#define BATCH 4
#define SEQ   4096

typedef float v2f __attribute__((ext_vector_type(2)));
typedef float v8f __attribute__((ext_vector_type(8)));

// ---------------------------------------------------------------------------
// Tiled GEMM on V_WMMA_F32_16X16X4_F32 (wave32, full f32 precision).
//   C[m,n] = alpha * sum_k A[m,k] * Bop[k,n] + (bias ? bias[n] : 0)
// B_IS_KXN = false : B stored [N,K] row-major (ldb = K)   (A·B^T, "NT")
// B_IS_KXN = true  : B stored [K,N] row-major (ldb = N)   (A·B,   "NN")
// Block tile 64(M) x 32(N), K staged in chunks of 32 through LDS.
// 256 threads = 8 waves; wave w owns 16x16 subtile at (w%4*16, w/4*16).
// ---------------------------------------------------------------------------
template <bool B_IS_KXN>
__global__ __launch_bounds__(256) void gemm_wmma_f32(
    const float* __restrict__ A, long batchStrideA, int lda,
    const float* __restrict__ B, long batchStrideB, int ldb,
    const float* __restrict__ bias,
    float* __restrict__ C, long batchStrideC, int ldc,
    int K, float alpha)
{
    __shared__ float As[64][36];   // 36-float row pitch: 16B-aligned, de-conflicted
    __shared__ float Bs[32][36];   // stored as Bs[k][n]

    const int tid  = threadIdx.x;
    const int wave = tid >> 5;     // wave32
    const int lane = tid & 31;
    const int half = lane >> 4;    // ISA 7.12.2: lanes 16..31 hold K+2 / M+8
    const int idx  = lane & 15;
    const int wm   = (wave & 3) * 16;
    const int wn   = (wave >> 2) * 16;

    const int bm = blockIdx.x * 64;
    const int bn = blockIdx.y * 32;
    const float* Ab = A + (size_t)blockIdx.z * batchStrideA;
    const float* Bb = B + (size_t)blockIdx.z * batchStrideB;
    float*       Cb = C + (size_t)blockIdx.z * batchStrideC;

    // A-tile loader: 8 floats/thread (2 x B128), row ar, cols ac..ac+7
    const int ar = tid >> 2;          // 0..63
    const int ac = (tid & 3) * 8;     // 0,8,16,24
    // B-tile loader: 4 floats/thread (1 x B128)
    const int br = tid >> 3;          // 0..31
    const int bc = (tid & 7) * 4;     // 0..28

    v8f acc = {0.f, 0.f, 0.f, 0.f, 0.f, 0.f, 0.f, 0.f};

    for (int k0 = 0; k0 < K; k0 += 32) {
        // ---- stage A tile: rows bm..bm+63, cols k0..k0+31 ----
        {
            const float4 a0 = *(const float4*)&Ab[(size_t)(bm + ar) * lda + (k0 + ac)];
            const float4 a1 = *(const float4*)&Ab[(size_t)(bm + ar) * lda + (k0 + ac + 4)];
            *(float4*)&As[ar][ac]     = a0;
            *(float4*)&As[ar][ac + 4] = a1;
        }
        // ---- stage B tile into Bs[k][n] ----
        if constexpr (!B_IS_KXN) {
            // B is [N,K]: read row (bn+br), K-fast; transpose into Bs
            const float4 bv4 = *(const float4*)&Bb[(size_t)(bn + br) * ldb + (k0 + bc)];
            Bs[bc + 0][br] = bv4.x;
            Bs[bc + 1][br] = bv4.y;
            Bs[bc + 2][br] = bv4.z;
            Bs[bc + 3][br] = bv4.w;
        } else {
            // B is [K,N]: read row (k0+br), N-fast; store directly
            const float4 bv4 = *(const float4*)&Bb[(size_t)(k0 + br) * ldb + (bn + bc)];
            *(float4*)&Bs[br][bc] = bv4;
        }
        __syncthreads();

        // ---- 8 x V_WMMA_F32_16X16X4_F32 per stage ----
#pragma unroll
        for (int kk = 0; kk < 32; kk += 4) {
            const int ks = kk + 2 * half;
            v2f a, b;
            a.x = As[wm + idx][ks];        // A 16x4: lane row = lane%16, K pair by half
            a.y = As[wm + idx][ks + 1];
            b.x = Bs[ks][wn + idx];        // B 4x16: lane col = lane%16, K pair by half
            b.y = Bs[ks + 1][wn + idx];
            acc = __builtin_amdgcn_wmma_f32_16x16x4_f32(
                false, a, false, b, (short)0, acc, false, false);
        }
        __syncthreads();
    }

    // ---- store: C/D 16x16 f32 layout: VGPR j -> row j + 8*half, col = lane%16 ----
    const int gn   = bn + wn + idx;
    const float bb = bias ? bias[gn] : 0.0f;
    const int  m0  = bm + wm + half * 8;
#pragma unroll
    for (int j = 0; j < 8; ++j) {
        Cb[(size_t)(m0 + j) * ldc + gn] = alpha * acc[j] + bb;
    }
}

// ---------------------------------------------------------------------------
// Softmax over the QUERY axis (dim=1): per column k, online max & sum-exp.
// Block = 256 threads handles 64 columns; 4-way row split merged via LDS.
// ---------------------------------------------------------------------------
__global__ __launch_bounds__(256) void softmax_col_stats(
    const float* __restrict__ attn, float* __restrict__ mcol, float* __restrict__ linv)
{
    __shared__ float sm[4][64];
    __shared__ float sl[4][64];
    const int tid = threadIdx.x;
    const int qp  = tid >> 6;                    // 0..3 row partition
    const int cl  = tid & 63;                    // column within block
    const int col = blockIdx.x * 64 + cl;        // 0 .. BATCH*SEQ-1
    const int b   = col >> 12;                   // /SEQ
    const int k   = col & (SEQ - 1);
    const float* p = attn + (size_t)b * SEQ * SEQ + k;

    float m = -INFINITY, l = 0.0f;
    for (int q = qp; q < SEQ; q += 4) {
        const float s  = p[(size_t)q * SEQ];
        const float mn = fmaxf(m, s);
        l = l * __expf(m - mn) + __expf(s - mn);
        m = mn;
    }
    sm[qp][cl] = m;
    sl[qp][cl] = l;
    __syncthreads();
    if (tid < 64) {
        float M = sm[0][tid], L = sl[0][tid];
#pragma unroll
        for (int i = 1; i < 4; ++i) {
            const float mi = sm[i][tid], li = sl[i][tid];
            const float mn = fmaxf(M, mi);
            L = L * __expf(M - mn) + li * __expf(mi - mn);
            M = mn;
        }
        mcol[blockIdx.x * 64 + tid] = M;
        linv[blockIdx.x * 64 + tid] = 1.0f / L;
    }
}

// w[q,k] = exp(s[q,k] - m[k]) * (1/l[k]), in place, float4 per thread.
__global__ __launch_bounds__(256) void softmax_col_norm(
    float* __restrict__ attn, const float* __restrict__ mcol, const float* __restrict__ linv)
{
    const size_t base = ((size_t)blockIdx.x * 256 + threadIdx.x) * 4;
    const int b = (int)(base >> 24);             // / (SEQ*SEQ) = 2^24
    const int k = (int)(base & (SEQ - 1));
    float4 s = *(float4*)&attn[base];
    const float* mp = mcol + (b << 12) + k;
    const float* lp = linv + (b << 12) + k;
    s.x = __expf(s.x - mp[0]) * lp[0];
    s.y = __expf(s.y - mp[1]) * lp[1];
    s.z = __expf(s.z - mp[2]) * lp[2];
    s.w = __expf(s.w - mp[3]) * lp[3];
    *(float4*)&attn[base] = s;
}

// ---------------------------------------------------------------------------
extern "C" void kernel_launch(void* const* d_in, const int* in_sizes, int n_in,
                              void* d_out, int out_size, void* d_ws, size_t ws_size,
                              hipStream_t stream)
{
    (void)in_sizes; (void)n_in; (void)out_size; (void)ws_size;

    const float* q_in = (const float*)d_in[0];
    const float* k_in = (const float*)d_in[1];
    const float* v_in = (const float*)d_in[2];
    const float* Wq   = (const float*)d_in[3];
    const float* bq   = (const float*)d_in[4];
    const float* Wk   = (const float*)d_in[5];
    const float* bk   = (const float*)d_in[6];
    const float* Wv   = (const float*)d_in[7];
    const float* bv   = (const float*)d_in[8];

    const long sxe = (long)SEQ * EMBED;          // 1,048,576
    const long sxs = (long)SEQ * SEQ;            // 16,777,216

    float* out  = (float*)d_out;                 // [B,S,E]
    float* attn = out + (size_t)BATCH * sxe;     // [B,S,S] (raw scores -> weights)

    float* Q    = (float*)d_ws;                  // [B,S,E]
    float* Km   = Q  + (size_t)BATCH * sxe;      // [B,S,E]
    float* Vm   = Km + (size_t)BATCH * sxe;      // [B,S,E]
    float* mcol = Vm + (size_t)BATCH * sxe;      // [B,S]
    float* linv = mcol + (size_t)BATCH * SEQ;    // [B,S]

    const dim3 blk(256);

    // Projections (NT). Reference swaps inputs: K <- Wk(v_input), V <- Wv(k_input).
    const dim3 gP(SEQ / 64, EMBED / 32, BATCH);
    gemm_wmma_f32<false><<<gP, blk, 0, stream>>>(q_in, sxe, EMBED, Wq, 0, EMBED, bq, Q,  sxe, EMBED, EMBED, 1.0f);
    gemm_wmma_f32<false><<<gP, blk, 0, stream>>>(v_in, sxe, EMBED, Wk, 0, EMBED, bk, Km, sxe, EMBED, EMBED, 1.0f);
    gemm_wmma_f32<false><<<gP, blk, 0, stream>>>(k_in, sxe, EMBED, Wv, 0, EMBED, bv, Vm, sxe, EMBED, EMBED, 1.0f);

    // scores = Q K^T / sqrt(256)  (NT), raw scores straight into d_out attn region
    const dim3 gS(SEQ / 64, SEQ / 32, BATCH);
    gemm_wmma_f32<false><<<gS, blk, 0, stream>>>(Q, sxe, EMBED, Km, sxe, EMBED, nullptr,
                                                 attn, sxs, SEQ, EMBED, 1.0f / 16.0f);

    // softmax over queries (column softmax), in place
    softmax_col_stats<<<dim3(BATCH * SEQ / 64), blk, 0, stream>>>(attn, mcol, linv);
    softmax_col_norm <<<dim3((unsigned)((BATCH * sxs) / 1024)), blk, 0, stream>>>(attn, mcol, linv);

    // out = attn @ V  (NN)
    const dim3 gO(SEQ / 64, EMBED / 32, BATCH);
    gemm_wmma_f32<true><<<gO, blk, 0, stream>>>(attn, sxs, SEQ, Vm, sxe, EMBED, nullptr,
                                                out, sxe, EMBED, SEQ, 1.0f);
}